// TimeAwareMultiHeadAttention_88682484728384
// MI455X (gfx1250) — compile-verified
//
#include <hip/hip_runtime.h>
#include <hip/hip_bf16.h>
#include <cstdint>

// ---------------------------------------------------------------------------
// TimeAwareMultiHeadAttention (TiSASRec) for MI455X / gfx1250.
//
// Roofline: 4x [B,L,L,H] fp32 tensors = ~655MB streamed once -> ~28us @ 23.3TB/s.
// FLOPs < 1 GFLOP -> memory bound. Keep fp32; use V_WMMA_F32_16X16X4_F32 for the
// two matmul-shaped pieces (Q(K+pK)^T and A(V+pV)) so matrix math rides the
// CDNA5 matrix pipe at full precision.
// ---------------------------------------------------------------------------

#define Bc   16
#define Lc   200
#define Hc   64
#define Dc   32      // head dim
#define NQT  13      // ceil(200/16)
#define LPAD 208     // 13*16
#define NEGV (-4294967295.0f)   // -2^32 + 1
#define SCALE 0.17677669529663688f  // 1/sqrt(32)

typedef __attribute__((ext_vector_type(2))) float v2f;
typedef __attribute__((ext_vector_type(8))) float v8f;

// ---------------------------------------------------------------------------
// Kernel 1: projections.  Q = q@Wq^T+bq ; KpK = k@Wk^T+bk+abs_pos_K ;
// VpV = k@Wv^T+bv+abs_pos_V.  Output laid out per-head: [h][B*L][32].
// ---------------------------------------------------------------------------
__global__ __launch_bounds__(256) void proj_kernel(
    const float* __restrict__ queries, const float* __restrict__ keys,
    const float* __restrict__ posK, const float* __restrict__ posV,
    const float* __restrict__ Wq, const float* __restrict__ bq,
    const float* __restrict__ Wk, const float* __restrict__ bk,
    const float* __restrict__ Wv, const float* __restrict__ bv,
    float* __restrict__ Qh, float* __restrict__ KpKh, float* __restrict__ VpVh)
{
    int gid = blockIdx.x * blockDim.x + threadIdx.x;   // [3][B*L][64]
    if (gid >= 3 * Bc * Lc * Hc) return;
    int col = gid & 63;
    int row = (gid >> 6) % (Bc * Lc);
    int p   = gid / ((Bc * Lc) * Hc);

    const float* in = (p == 0) ? queries : keys;
    const float* W  = (p == 0) ? Wq : (p == 1 ? Wk : Wv);
    const float* bb = (p == 0) ? bq : (p == 1 ? bk : bv);

    const float* inr = in + (size_t)row * Hc;
    const float* wr  = W  + (size_t)col * Hc;
    float acc = bb[col];
#pragma unroll 8
    for (int i = 0; i < Hc; ++i) acc += inr[i] * wr[i];
    if (p == 1) acc += posK[(size_t)row * Hc + col];
    if (p == 2) acc += posV[(size_t)row * Hc + col];

    int h = col >> 5, dd = col & 31;
    float* dst = (p == 0) ? Qh : (p == 1 ? KpKh : VpVh);
    dst[((size_t)h * (Bc * Lc) + row) * Dc + dd] = acc;
}

// ---------------------------------------------------------------------------
// Kernel 2: one block per (qtile, batch, head).  256 threads = 8 wave32s.
// ---------------------------------------------------------------------------
__global__ __launch_bounds__(256) void attn_kernel(
    const float* __restrict__ Qh, const float* __restrict__ KpKh,
    const float* __restrict__ VpVh,
    const float* __restrict__ tK, const float* __restrict__ dK,
    const float* __restrict__ tV, const float* __restrict__ dV,
    const unsigned char* __restrict__ timeMask,
    const unsigned char* __restrict__ attnMask,
    float* __restrict__ out)
{
    __shared__ float Qs[16][Dc];        // query tile
    __shared__ float Ks[LPAD][Dc];      // K+pK (rows 200..207 zero)
    __shared__ float Vs[LPAD][Dc];      // V+pV (rows 200..207 zero)
    __shared__ float S[16][LPAD];       // scores -> unnormalized softmax A
    __shared__ float rowsum[16];
    __shared__ float O[16][Dc];

    const int tid  = threadIdx.x;
    const int w    = tid >> 5;          // wave id 0..7
    const int lane = tid & 31;
    const int half = lane >> 4;         // 0|1
    const int m16  = lane & 15;

    const int qt = blockIdx.x;          // 0..12
    const int b  = blockIdx.y;          // 0..15
    const int h  = blockIdx.z;          // 0..1
    const int q0 = qt * 16;

    const size_t headOff = ((size_t)h * (Bc * Lc) + (size_t)b * Lc) * Dc;
    // base into [B,L,L,H] tensors for this (b, head): + (gq*L + k)*H later
    const size_t bigBase = ((size_t)b * Lc) * Lc * Hc + (size_t)h * Dc;

    // ---- stage 0: fill LDS -------------------------------------------------
    for (int i = tid; i < 16 * Dc; i += 256) {
        int q = i >> 5, dd = i & 31, gq = q0 + q;
        Qs[q][dd] = (gq < Lc) ? Qh[headOff + (size_t)gq * Dc + dd] : 0.0f;
        O[q][dd]  = 0.0f;
    }
    for (int i = tid; i < (Lc * Dc) / 4; i += 256) {   // 1600 float4
        ((float4*)Ks)[i] = ((const float4*)(KpKh + headOff))[i];
        ((float4*)Vs)[i] = ((const float4*)(VpVh + headOff))[i];
    }
    for (int i = tid; i < 8 * Dc; i += 256) {          // zero pad rows 200..207
        int r = Lc + (i >> 5), dd = i & 31;
        Ks[r][dd] = 0.0f;
        Vs[r][dd] = 0.0f;
    }
    if (tid < 16) rowsum[tid] = 1.0f;
    __syncthreads();

    // ---- stage 1: S1 = Q_tile @ (K+pK)^T via V_WMMA_F32_16X16X4_F32 --------
    // A(16x4 f32): lane l holds row m16, K = 2*half + r  (ISA 7.12.2)
    // B(4x16 f32): lane l holds K = 2*half + r, col m16
    // C/D(16x16):  reg r -> row r + 8*half, col m16
    v2f aQ[8];
#pragma unroll
    for (int c = 0; c < 8; ++c) {
        int db = 4 * c + 2 * half;
        aQ[c].x = Qs[m16][db];
        aQ[c].y = Qs[m16][db + 1];
    }
    for (int kt = w; kt < NQT; kt += 8) {   // uniform per wave -> EXEC all ones
        v8f acc = {};
#pragma unroll
        for (int c = 0; c < 8; ++c) {
            int db = 4 * c + 2 * half;
            v2f bK;
            bK.x = Ks[kt * 16 + m16][db];
            bK.y = Ks[kt * 16 + m16][db + 1];
            acc = __builtin_amdgcn_wmma_f32_16x16x4_f32(
                false, aQ[c], false, bK, (short)0, acc, false, false);
        }
#pragma unroll
        for (int r = 0; r < 8; ++r)
            S[r + 8 * half][kt * 16 + m16] = acc[r];
    }
    __syncthreads();

    // ---- stage 2: S += (tK+dK)·Q, scale, mask (streams 2x164MB total) ------
    for (int idx = tid; idx < 16 * Lc; idx += 256) {
        int q = idx / Lc, k = idx - q * Lc;
        int gq = q0 + q;
        if (gq >= Lc) continue;
        const float4* tkp = (const float4*)(tK + bigBase + ((size_t)gq * Lc + k) * Hc);
        const float4* dkp = (const float4*)(dK + bigBase + ((size_t)gq * Lc + k) * Hc);
        float acc = 0.0f;
#pragma unroll
        for (int j = 0; j < 8; ++j) {
            float4 a4 = tkp[j], c4 = dkp[j];
            acc += (a4.x + c4.x) * Qs[q][4 * j + 0]
                 + (a4.y + c4.y) * Qs[q][4 * j + 1]
                 + (a4.z + c4.z) * Qs[q][4 * j + 2]
                 + (a4.w + c4.w) * Qs[q][4 * j + 3];
        }
        float v = (S[q][k] + acc) * SCALE;
        bool mk = (timeMask[b * Lc + gq] != 0) || (attnMask[(size_t)gq * Lc + k] != 0);
        S[q][k] = mk ? NEGV : v;
    }
    __syncthreads();

    // ---- stage 3: row softmax (unnormalized A in S, 1/sum deferred) --------
    for (int r = w * 2; r < w * 2 + 2; ++r) {
        float mx = -3.4e38f;
        for (int k = lane; k < Lc; k += 32) mx = fmaxf(mx, S[r][k]);
#pragma unroll
        for (int off = 16; off >= 1; off >>= 1) mx = fmaxf(mx, __shfl_xor(mx, off, 32));
        float sum = 0.0f;
        for (int k = lane; k < LPAD; k += 32) {
            float e = (k < Lc) ? __expf(S[r][k] - mx) : 0.0f;
            S[r][k] = e;
            sum += e;
        }
#pragma unroll
        for (int off = 16; off >= 1; off >>= 1) sum += __shfl_xor(sum, off, 32);
        if (lane == 0) rowsum[r] = sum;
    }
    __syncthreads();

    // ---- stage 4: O1 = A @ (V+pV) via WMMA; 8 waves = 2 n-tiles x 4 K-segs -
    {
        int nt  = w >> 2;               // d-column tile 0..1
        int seg = w & 3;                // K segment 0..3 (13 chunks each)
        v8f acc = {};
        for (int i = 0; i < 13; ++i) {
            int k0 = (seg * 13 + i) * 4;
            v2f aA, bV;
            aA.x = S[m16][k0 + 2 * half];
            aA.y = S[m16][k0 + 2 * half + 1];
            bV.x = Vs[k0 + 2 * half][nt * 16 + m16];
            bV.y = Vs[k0 + 2 * half + 1][nt * 16 + m16];
            acc = __builtin_amdgcn_wmma_f32_16x16x4_f32(
                false, aA, false, bV, (short)0, acc, false, false);
        }
#pragma unroll
        for (int r = 0; r < 8; ++r)
            atomicAdd(&O[r + 8 * half][nt * 16 + m16], acc[r]);   // ds_add_f32
    }

    // ---- stage 5: O += sum_k A[q,k]*(tV+dV)[q,k,:]  (streams 2x164MB) ------
    // lane = channel -> perfectly coalesced 128B loads per wave instruction.
    for (int rr = 0; rr < 2; ++rr) {
        int q = w * 2 + rr, gq = q0 + q;
        if (gq >= Lc) continue;
        const float* tvp = tV + bigBase + (size_t)gq * Lc * Hc + lane;
        const float* dvp = dV + bigBase + (size_t)gq * Lc * Hc + lane;
        float acc = 0.0f;
        for (int k = 0; k < Lc; ++k) {
            float a = S[q][k];
            acc += a * (tvp[(size_t)k * Hc] + dvp[(size_t)k * Hc]);
        }
        atomicAdd(&O[q][lane], acc);
    }
    __syncthreads();

    // ---- stage 6: normalize + store ----------------------------------------
    for (int i = tid; i < 16 * Dc; i += 256) {
        int q = i >> 5, dd = i & 31, gq = q0 + q;
        if (gq < Lc)
            out[((size_t)b * Lc + gq) * Hc + h * Dc + dd] = O[q][dd] / rowsum[q];
    }
}

// ---------------------------------------------------------------------------
extern "C" void kernel_launch(void* const* d_in, const int* in_sizes, int n_in,
                              void* d_out, int out_size, void* d_ws, size_t ws_size,
                              hipStream_t stream)
{
    (void)in_sizes; (void)n_in; (void)out_size; (void)ws_size;
    const float* queries = (const float*)d_in[0];
    const float* keys    = (const float*)d_in[1];
    const unsigned char* time_mask = (const unsigned char*)d_in[2];
    const unsigned char* attn_mask = (const unsigned char*)d_in[3];
    const float* tK = (const float*)d_in[4];
    const float* tV = (const float*)d_in[5];
    const float* dKm = (const float*)d_in[6];
    const float* dVm = (const float*)d_in[7];
    const float* pK = (const float*)d_in[8];
    const float* pV = (const float*)d_in[9];
    const float* Wq = (const float*)d_in[10];
    const float* bq = (const float*)d_in[11];
    const float* Wk = (const float*)d_in[12];
    const float* bk = (const float*)d_in[13];
    const float* Wv = (const float*)d_in[14];
    const float* bv = (const float*)d_in[15];

    // workspace: 3 x [2][16*200][32] fp32 = 3 x 819.2 KB
    float* ws   = (float*)d_ws;
    float* Qh   = ws;
    float* KpKh = ws + 204800;
    float* VpVh = ws + 409600;

    int projN = 3 * Bc * Lc * Hc;   // 614400
    proj_kernel<<<dim3((projN + 255) / 256), 256, 0, stream>>>(
        queries, keys, pK, pV, Wq, bq, Wk, bk, Wv, bv, Qh, KpKh, VpVh);

    attn_kernel<<<dim3(NQT, Bc, 2), 256, 0, stream>>>(
        Qh, KpKh, VpVh, tK, dKm, tV, dVm, time_mask, attn_mask, (float*)d_out);
}